// GroverMPNEncoder_33947421507739
// MI455X (gfx1250) — compile-verified
//
#include <hip/hip_runtime.h>
#include <hip/hip_bf16.h>

#define HIDDEN 128
#define DEPTH  6

typedef __attribute__((ext_vector_type(2))) float v2f;
typedef __attribute__((ext_vector_type(8))) float v8f;

// ---------------------------------------------------------------------------
// Zero a float4-aligned buffer (agg / deg). Grid-stride.
// ---------------------------------------------------------------------------
__global__ void zero4_kernel(float4* __restrict__ p, long long n4) {
    long long i = (long long)blockIdx.x * blockDim.x + threadIdx.x;
    long long stride = (long long)gridDim.x * blockDim.x;
    float4 z = {0.f, 0.f, 0.f, 0.f};
    for (; i < n4; i += stride) p[i] = z;
}

// ---------------------------------------------------------------------------
// In-degree via fp32 atomics: deg[col[e]] += 1
// ---------------------------------------------------------------------------
__global__ void deg_kernel(const int* __restrict__ col, float* __restrict__ deg,
                           int n_edges) {
    int e = blockIdx.x * blockDim.x + threadIdx.x;
    if (e < n_edges) atomicAdd(&deg[col[e]], 1.0f);
}

// deg[i] = 1 / max(deg[i], 1)  (precompute reciprocal once; mul beats div later)
__global__ void rdeg_kernel(float* __restrict__ deg, int n_nodes) {
    int i = blockIdx.x * blockDim.x + threadIdx.x;
    if (i < n_nodes) deg[i] = 1.0f / fmaxf(deg[i], 1.0f);
}

// ---------------------------------------------------------------------------
// Edge scatter: one wave32 per edge. Lane l owns features [4l, 4l+4).
// x row (512B) gathered as float4 (L2-resident), scattered with
// global_atomic_add_f32 into agg[col].
// ---------------------------------------------------------------------------
__global__ void __launch_bounds__(256)
scatter_kernel(const float* __restrict__ x, const int* __restrict__ row,
               const int* __restrict__ col, float* __restrict__ agg, int n_edges) {
    int wave = (int)((blockIdx.x * blockDim.x + threadIdx.x) >> 5);
    int lane = threadIdx.x & 31;
    if (wave >= n_edges) return;
    int r = row[wave];   // uniform across the wave -> scalarized by compiler
    int c = col[wave];
    float4 v = ((const float4*)(x + (long long)r * HIDDEN))[lane];
    float* dst = agg + (long long)c * HIDDEN + lane * 4;
    atomicAdd(dst + 0, v.x);
    atomicAdd(dst + 1, v.y);
    atomicAdd(dst + 2, v.z);
    atomicAdd(dst + 3, v.w);
}

// ---------------------------------------------------------------------------
// Fused: x += relu( (agg * rdeg) @ W^T + bias )
// 256 threads = 8 waves. Block handles 16-node tiles (grid-stride).
// Wave w owns output columns [16w, 16w+16); preloads its 32 B-fragments of
// W^T into registers once. A comes from an LDS-staged, rdeg-scaled 16x128
// tile. 32x V_WMMA_F32_16X16X4_F32 per tile.
// ---------------------------------------------------------------------------
__global__ void __launch_bounds__(256)
gemm_kernel(const float* __restrict__ agg, const float* __restrict__ rdeg,
            const float* __restrict__ W, const float* __restrict__ bias,
            float* __restrict__ x, int n_nodes) {
    __shared__ float sIn[16][HIDDEN + 4];   // +4 pad: row stride 132 -> no 16-way bank conflict

    const int tid    = threadIdx.x;
    const int waveId = tid >> 5;            // 0..7
    const int lane   = tid & 31;
    const int half   = lane >> 4;           // 0: lanes 0-15, 1: lanes 16-31
    const int l16    = lane & 15;
    const int n      = waveId * 16 + l16;   // output feature owned by this lane

    // B-fragment s (k0 = 4s): VGPR v holds B[k0 + v + 2*half][n] = W[n*128 + k0 + v + 2*half]
    v2f bfrag[32];
    {
        const float* wrow = W + (long long)n * HIDDEN + 2 * half;
        #pragma unroll
        for (int s = 0; s < 32; ++s) {
            bfrag[s].x = wrow[4 * s + 0];
            bfrag[s].y = wrow[4 * s + 1];
        }
    }
    const float bval = bias[n];

    const int n_tiles = (n_nodes + 15) >> 4;
    for (int tile = blockIdx.x; tile < n_tiles; tile += gridDim.x) {
        const int row0 = tile << 4;

        __syncthreads();   // previous iteration's reads of sIn must finish
        // Stage 16x128 tile, scaled by rdeg: 512 float4, 2 per thread.
        for (int i = tid; i < 16 * (HIDDEN / 4); i += 256) {
            int m    = i >> 5;       // 0..15
            int kq   = i & 31;       // float4 index in row
            int node = row0 + m;
            float4 v = {0.f, 0.f, 0.f, 0.f};
            float  s = 0.f;
            if (node < n_nodes) {
                v = ((const float4*)(agg + (long long)node * HIDDEN))[kq];
                s = rdeg[node];
            }
            sIn[m][kq * 4 + 0] = v.x * s;
            sIn[m][kq * 4 + 1] = v.y * s;
            sIn[m][kq * 4 + 2] = v.z * s;
            sIn[m][kq * 4 + 3] = v.w * s;
        }
        __syncthreads();

        v8f c = {};   // zero accumulator
        #pragma unroll
        for (int s = 0; s < 32; ++s) {
            const int k0 = 4 * s;
            v2f a;
            a.x = sIn[l16][k0 + 2 * half + 0];   // A[l16][k0 + v + 2*half]
            a.y = sIn[l16][k0 + 2 * half + 1];
            c = __builtin_amdgcn_wmma_f32_16x16x4_f32(
                    /*neg_a=*/false, a, /*neg_b=*/false, bfrag[s],
                    /*c_mod=*/(short)0, c, /*reuse_a=*/false, /*reuse_b=*/false);
        }

        // Epilogue: VGPR v of C/D holds (M = v + 8*half, N = n).
        #pragma unroll
        for (int v = 0; v < 8; ++v) {
            int node = row0 + v + 8 * half;
            if (node < n_nodes) {
                float* px  = x + (long long)node * HIDDEN + n;
                float  val = fmaxf(c[v] + bval, 0.f);
                *px = *px + val;
            }
        }
    }
}

// ---------------------------------------------------------------------------
// Launch: deg once, then DEPTH x (zero agg -> scatter -> fused WMMA GEMM).
// ---------------------------------------------------------------------------
extern "C" void kernel_launch(void* const* d_in, const int* in_sizes, int n_in,
                              void* d_out, int out_size, void* d_ws, size_t ws_size,
                              hipStream_t stream) {
    const float* x_in  = (const float*)d_in[0];
    const int*   edges = (const int*)d_in[1];     // [2, E]
    const float* W     = (const float*)d_in[2];
    const float* bias  = (const float*)d_in[3];

    const int n_nodes = in_sizes[0] / HIDDEN;
    const int n_edges = in_sizes[1] / 2;
    const int* row = edges;
    const int* col = edges + n_edges;

    float* x   = (float*)d_out;                               // running node state
    float* agg = (float*)d_ws;                                // [N, 128]
    float* deg = agg + (long long)n_nodes * HIDDEN;           // [N] -> becomes rdeg

    const long long nxf = (long long)n_nodes * HIDDEN;

    // x <- x_in (device-to-device, capture-safe)
    hipMemcpyAsync(x, x_in, (size_t)nxf * sizeof(float),
                   hipMemcpyDeviceToDevice, stream);

    // reciprocal clipped degree (constant across depth)
    {
        long long n4 = (long long)n_nodes / 4;
        int blocks = (int)((n4 + 255) / 256);
        zero4_kernel<<<blocks, 256, 0, stream>>>((float4*)deg, n4);
        deg_kernel<<<(n_edges + 255) / 256, 256, 0, stream>>>(col, deg, n_edges);
        rdeg_kernel<<<(n_nodes + 255) / 256, 256, 0, stream>>>(deg, n_nodes);
    }

    const long long agg4      = nxf / 4;
    const int zero_blocks     = (int)((agg4 + 255) / 256) < 16384
                                ? (int)((agg4 + 255) / 256) : 16384;
    const int scatter_blocks  = (int)(((long long)n_edges * 32 + 255) / 256);
    const int n_tiles         = (n_nodes + 15) >> 4;
    const int gemm_blocks     = n_tiles < 3200 ? n_tiles : 3200;

    for (int d = 0; d < DEPTH; ++d) {
        zero4_kernel<<<zero_blocks, 256, 0, stream>>>((float4*)agg, agg4);
        scatter_kernel<<<scatter_blocks, 256, 0, stream>>>(x, row, col, agg, n_edges);
        gemm_kernel<<<gemm_blocks, 256, 0, stream>>>(agg, deg, W, bias, x, n_nodes);
    }
}